// Snook_27668179321127
// MI455X (gfx1250) — compile-verified
//
#include <hip/hip_runtime.h>
#include <hip/hip_bf16.h>
#include <math.h>

// ---------------------------------------------------------------------------
// Pipeline: 4K image -> (256,64) antialiased resizes -> 3x3 convs ->
// heatmap -> ordered peak compaction -> pairwise reject -> window gather ->
// feats(256x3072) @ W(3072x7) via V_WMMA_F32_16X16X4_F32 -> softmax/argmax.
// ---------------------------------------------------------------------------

#define FILL_I   (-1000000)
#define FILL_F   (-1000000.0f)
#define HM       256
#define K_MAX    256
#define NCLS     7
#define INH      2160
#define INW      3840

typedef float v2f __attribute__((ext_vector_type(2)));
typedef float v8f __attribute__((ext_vector_type(8)));

// ---- output layout (floats) ----
#define OUT_HEAT    0          // 256*256
#define OUT_COORDS  65536      // 256*2
#define OUT_SURV    66048      // 256
#define OUT_LOGITS  66304      // 256*7
#define OUT_LABELS  68096      // 256

// ---- workspace layout (float offsets) ----
#define O_TH      0            // 3*2160*256 = 1658880
#define O_TM      1658880      // 3*2160*64  = 414720
#define O_HX      2073600      // 3*256*256  = 196608
#define O_MX      2270208      // 3*64*64    = 12288
#define O_MK      2282496      // 64*64      = 4096
#define O_WPAD    2286592      // 3072*16    = 49152
#define O_FEATS   2335744      // 256*3072   = 786432
#define O_LOGP    3122176      // 256*16     = 4096
#define O_COORDI  3126272      // 256*2 ints (as float slots)
// total ~ 3126784 floats = 12.5 MB

// ---------------------------------------------------------------------------
// Horizontal antialiased triangle resize: in width 3840 -> outW, integer scale S.
// sample*2 = (2o+1)*S - 1 (exact). weight = max(0, 1 - |U - 2j|/(2S)).
// ---------------------------------------------------------------------------
__global__ __launch_bounds__(256)
void hresize_kernel(const float* __restrict__ x, float* __restrict__ out,
                    int outW, int S) {
  int idx = blockIdx.x * blockDim.x + threadIdx.x;
  int total = 3 * INH * outW;
  if (idx >= total) return;
  int o   = idx % outW;
  int row = idx / outW;                 // c*2160 + y
  const float* xr = x + (size_t)row * INW;
  int U = (2 * o + 1) * S - 1;
  int jlo = (U - 2 * S) / 2 - 1; if (jlo < 0) jlo = 0;
  int jhi = (U + 2 * S) / 2 + 1; if (jhi > INW - 1) jhi = INW - 1;
  float inv2S = 1.0f / (2.0f * (float)S);
  float acc = 0.0f, wsum = 0.0f;
  for (int j = jlo; j <= jhi; ++j) {
    float w = 1.0f - fabsf((float)(U - 2 * j)) * inv2S;
    if (w > 0.0f) { acc += w * xr[j]; wsum += w; }
  }
  out[idx] = acc / wsum;
}

// ---------------------------------------------------------------------------
// Vertical resize: 2160 rows -> outH rows, scale num/den (135/16 or 135/4).
// src layout (3, 2160, W), dst layout (3, outH, W).
// ---------------------------------------------------------------------------
__global__ __launch_bounds__(256)
void vresize_kernel(const float* __restrict__ src, float* __restrict__ dst,
                    int outH, int W, int num, int den) {
  int idx = blockIdx.x * blockDim.x + threadIdx.x;
  int total = 3 * outH * W;
  if (idx >= total) return;
  int w = idx % W;
  int t = idx / W;
  int o = t % outH;
  int c = t / outH;
  int U      = (2 * o + 1) * num - den;    // = 2*den*sample  (exact)
  int twoden = 2 * den;
  int jlo = (U - 2 * num) / twoden - 1; if (jlo < 0) jlo = 0;
  int jhi = (U + 2 * num) / twoden + 1; if (jhi > INH - 1) jhi = INH - 1;
  float inv2num = 1.0f / (2.0f * (float)num);
  const float* base = src + ((size_t)c * INH) * W + w;
  float acc = 0.0f, wsum = 0.0f;
  for (int j = jlo; j <= jhi; ++j) {
    float wt = 1.0f - fabsf((float)(U - twoden * j)) * inv2num;
    if (wt > 0.0f) { acc += wt * base[(size_t)j * W]; wsum += wt; }
  }
  dst[idx] = acc / wsum;
}

// 3-channel 3x3 cross-correlation, SAME zero pad. wt layout (1,3,3,3) OIHW.
__device__ __forceinline__ float conv3x3(const float* __restrict__ img, int H, int W,
                                         int r, int c, const float* __restrict__ wt) {
  float acc = 0.0f;
  #pragma unroll
  for (int ch = 0; ch < 3; ++ch) {
    const float* p = img + (size_t)ch * H * W;
    const float* k = wt + ch * 9;
    #pragma unroll
    for (int dy = 0; dy < 3; ++dy) {
      int y = r + dy - 1;
      if (y < 0 || y >= H) continue;
      #pragma unroll
      for (int dx = 0; dx < 3; ++dx) {
        int xx = c + dx - 1;
        if (xx < 0 || xx >= W) continue;
        acc += p[(size_t)y * W + xx] * k[dy * 3 + dx];
      }
    }
  }
  return acc;
}

__global__ __launch_bounds__(256)
void mask_kernel(const float* __restrict__ mx, const float* __restrict__ mw,
                 const float* __restrict__ mb, float* __restrict__ mk) {
  int idx = blockIdx.x * blockDim.x + threadIdx.x;
  if (idx >= 64 * 64) return;
  int r = idx >> 6, c = idx & 63;
  float v = conv3x3(mx, 64, 64, r, c, mw) + mb[0];
  v = fminf(fmaxf(v, 0.0f), 1.0f);
  mk[idx] = (v < 0.6f) ? 0.0f : v;
}

__global__ __launch_bounds__(256)
void heat_kernel(const float* __restrict__ hx, const float* __restrict__ lw,
                 const float* __restrict__ lb, const float* __restrict__ mk,
                 float* __restrict__ heat) {
  int idx = blockIdx.x * blockDim.x + threadIdx.x;
  if (idx >= HM * HM) return;
  int r = idx >> 8, c = idx & 255;
  float v = conv3x3(hx, HM, HM, r, c, lw) + lb[0];
  v = fminf(fmaxf(v, 0.0f), 1.0f);
  if (v < 0.4f) v = 0.0f;
  heat[idx] = v * mk[(r >> 2) * 64 + (c >> 2)];
}

// ---------------------------------------------------------------------------
// Peak detection + order-preserving compaction (row-major, first 256),
// single workgroup, 1024 threads x 64 contiguous cells, LDS scan.
// ---------------------------------------------------------------------------
__device__ __forceinline__ bool is_peak(const float* __restrict__ h, int cell) {
  int r = cell >> 8, c = cell & 255;
  float v  = h[cell];
  float lt = (c > 0)   ? h[cell - 1]   : 0.0f;
  float rt = (c < 255) ? h[cell + 1]   : 0.0f;
  float up = (r > 0)   ? h[cell - 256] : 0.0f;
  float dn = (r < 255) ? h[cell + 256] : 0.0f;
  return (v > lt) && (v > rt) && (v > up) && (v > dn);
}

__global__ __launch_bounds__(1024)
void peaks_kernel(const float* __restrict__ heat, int* __restrict__ coords_i,
                  float* __restrict__ coords_f) {
  __shared__ unsigned cnt[1024];
  int tid  = threadIdx.x;
  int base = tid * 64;
  unsigned my = 0;
  for (int k = 0; k < 64; ++k)
    if (is_peak(heat, base + k)) my++;
  cnt[tid] = my;
  __syncthreads();
  // inclusive Hillis-Steele scan
  for (int off = 1; off < 1024; off <<= 1) {
    unsigned v = (tid >= off) ? cnt[tid - off] : 0u;
    __syncthreads();
    cnt[tid] += v;
    __syncthreads();
  }
  unsigned pos   = cnt[tid] - my;   // exclusive prefix
  unsigned total = cnt[1023];
  for (int k = 0; k < 64; ++k) {
    int cell = base + k;
    if (is_peak(heat, cell)) {
      if (pos < K_MAX) {
        int r = cell >> 8, c = cell & 255;
        coords_i[2 * pos]     = r;
        coords_i[2 * pos + 1] = c;
        coords_f[2 * pos]     = (float)r;
        coords_f[2 * pos + 1] = (float)c;
      }
      pos++;
    }
  }
  if (tid < K_MAX && (unsigned)tid >= total) {
    coords_i[2 * tid]     = FILL_I;
    coords_i[2 * tid + 1] = FILL_I;
    coords_f[2 * tid]     = FILL_F;
    coords_f[2 * tid + 1] = FILL_F;
  }
}

__global__ __launch_bounds__(256)
void survive_kernel(const int* __restrict__ coords_i, float* __restrict__ surv) {
  __shared__ float rs[K_MAX], cs[K_MAX];
  __shared__ int   vi[K_MAX];
  int j = threadIdx.x;
  int rj = coords_i[2 * j], cj = coords_i[2 * j + 1];
  rs[j] = (float)rj; cs[j] = (float)cj; vi[j] = (rj > FILL_I) ? 1 : 0;
  __syncthreads();
  float fr = rs[j], fc = cs[j];
  bool reject = false;
  for (int i = 0; i < K_MAX; ++i) {
    if (!vi[i]) continue;
    float dr = rs[i] - fr, dc = cs[i] - fc;
    float d2 = dr * dr + dc * dc;
    if (d2 > 1.0f && d2 < 9.0f) { reject = true; break; }
  }
  surv[j] = (vi[j] && !reject) ? 1.0f : 0.0f;
}

// 32x32x3 windows from h_x around clipped coords (zero-pad 16) -> feats row-major
__global__ __launch_bounds__(256)
void gather_kernel(const float* __restrict__ hx, const int* __restrict__ coords_i,
                   float* __restrict__ feats) {
  int idx = blockIdx.x * blockDim.x + threadIdx.x;
  if (idx >= K_MAX * 3072) return;
  int e = idx % 3072, k = idx / 3072;
  int ch = e >> 10, rem = e & 1023, dy = rem >> 5, dx = rem & 31;
  int r = coords_i[2 * k];
  int c = coords_i[2 * k + 1];
  r = min(max(r, 0), HM - 1);
  c = min(max(c, 0), HM - 1);
  int y = r - 16 + dy, x = c - 16 + dx;
  float v = (y >= 0 && y < HM && x >= 0 && x < HM)
              ? hx[((size_t)ch * HM + y) * HM + x] : 0.0f;
  feats[idx] = v;
}

// pad label_w (3072x7) -> (3072x16)
__global__ __launch_bounds__(256)
void padw_kernel(const float* __restrict__ W, float* __restrict__ Wpad) {
  int idx = blockIdx.x * blockDim.x + threadIdx.x;
  if (idx >= 3072 * 16) return;
  int n = idx & 15, k = idx >> 4;
  Wpad[idx] = (n < NCLS) ? W[k * NCLS + n] : 0.0f;
}

// ---------------------------------------------------------------------------
// GEMM: feats(256x3072) @ Wpad(3072x16) + bias via V_WMMA_F32_16X16X4_F32.
// One wave per 16-row tile; 768 k-steps of K=4.
// A 16x4 layout: lane l -> row l&15, K = k0 + 2*(l>=16) + {0,1} (2 VGPRs).
// B  4x16 layout: lane l -> col l&15, K = k0 + 2*(l>=16) + {0,1}.
// C/D: VGPR v -> M = v + 8*(l>=16), N = l&15.
// ---------------------------------------------------------------------------
__global__ __launch_bounds__(32)
void gemm_wmma_kernel(const float* __restrict__ feats, const float* __restrict__ Wpad,
                      const float* __restrict__ bias, float* __restrict__ logits_pre) {
  int tile = blockIdx.x;          // 0..15
  int lane = threadIdx.x;         // 0..31
  int n = lane & 15;
  int h = lane >> 4;              // 0 or 1
  int row = tile * 16 + (lane & 15);
  float bn = (n < NCLS) ? bias[n] : 0.0f;
  v8f c;
  #pragma unroll
  for (int v = 0; v < 8; ++v) c[v] = bn;
  const float* arow = feats + (size_t)row * 3072;
  for (int k0 = 0; k0 < 3072; k0 += 4) {
    int ka = k0 + 2 * h;
    v2f a = *(const v2f*)(arow + ka);
    v2f b;
    b.x = Wpad[(size_t)ka * 16 + n];
    b.y = Wpad[(size_t)(ka + 1) * 16 + n];
    c = __builtin_amdgcn_wmma_f32_16x16x4_f32(false, a, false, b,
                                              (short)0, c, false, false);
  }
  int mbase = tile * 16 + (h << 3);
  #pragma unroll
  for (int v = 0; v < 8; ++v)
    logits_pre[(size_t)(mbase + v) * 16 + n] = c[v];
}

// softmax (== exp(log_softmax)) * survive, argmax -> labels
__global__ __launch_bounds__(256)
void finish_kernel(const float* __restrict__ logits_pre, const float* __restrict__ surv,
                   float* __restrict__ out_logits, float* __restrict__ out_labels) {
  int k = threadIdx.x;
  float l[NCLS];
  float mx = -INFINITY;
  #pragma unroll
  for (int n = 0; n < NCLS; ++n) { l[n] = logits_pre[k * 16 + n]; mx = fmaxf(mx, l[n]); }
  float s = 0.0f;
  #pragma unroll
  for (int n = 0; n < NCLS; ++n) { l[n] = expf(l[n] - mx); s += l[n]; }
  float scale = surv[k] / s;
  float best = -1.0f; int bi = 0;
  #pragma unroll
  for (int n = 0; n < NCLS; ++n) {
    float p = l[n] * scale;
    out_logits[k * NCLS + n] = p;
    if (p > best) { best = p; bi = n; }
  }
  out_labels[k] = (float)bi;
}

extern "C" void kernel_launch(void* const* d_in, const int* in_sizes, int n_in,
                              void* d_out, int out_size, void* d_ws, size_t ws_size,
                              hipStream_t stream) {
  (void)in_sizes; (void)n_in; (void)out_size; (void)ws_size;
  const float* x       = (const float*)d_in[0];
  const float* loc_w   = (const float*)d_in[1];
  const float* loc_b   = (const float*)d_in[2];
  const float* mask_w  = (const float*)d_in[3];
  const float* mask_b  = (const float*)d_in[4];
  const float* label_w = (const float*)d_in[5];
  const float* label_b = (const float*)d_in[6];

  float* out = (float*)d_out;
  float* ws  = (float*)d_ws;

  float* t_h      = ws + O_TH;
  float* t_m      = ws + O_TM;
  float* h_x      = ws + O_HX;
  float* m_x      = ws + O_MX;
  float* mk       = ws + O_MK;
  float* wpad     = ws + O_WPAD;
  float* feats    = ws + O_FEATS;
  float* logp     = ws + O_LOGP;
  int*   coords_i = (int*)(ws + O_COORDI);

  // 1) horizontal resizes (read 4K input once; second pass hits L2)
  hresize_kernel<<<(3 * INH * 256 + 255) / 256, 256, 0, stream>>>(x, t_h, 256, 15);
  hresize_kernel<<<(3 * INH * 64  + 255) / 256, 256, 0, stream>>>(x, t_m, 64, 60);
  // 2) vertical resizes (2160 -> 256 : S=135/16 ; 2160 -> 64 : S=135/4)
  vresize_kernel<<<(3 * 256 * 256 + 255) / 256, 256, 0, stream>>>(t_h, h_x, 256, 256, 135, 16);
  vresize_kernel<<<(3 * 64 * 64   + 255) / 256, 256, 0, stream>>>(t_m, m_x, 64, 64, 135, 4);
  // 3) mask conv + heatmap (heatmap goes straight to d_out)
  mask_kernel<<<16, 256, 0, stream>>>(m_x, mask_w, mask_b, mk);
  heat_kernel<<<256, 256, 0, stream>>>(h_x, loc_w, loc_b, mk, out + OUT_HEAT);
  // 4) ordered peak compaction
  peaks_kernel<<<1, 1024, 0, stream>>>(out + OUT_HEAT, coords_i, out + OUT_COORDS);
  // 5) pairwise reject
  survive_kernel<<<1, 256, 0, stream>>>(coords_i, out + OUT_SURV);
  // 6) window gather + weight pad
  gather_kernel<<<(K_MAX * 3072 + 255) / 256, 256, 0, stream>>>(h_x, coords_i, feats);
  padw_kernel<<<(3072 * 16 + 255) / 256, 256, 0, stream>>>(label_w, wpad);
  // 7) WMMA GEMM (f32 16x16x4), one wave per 16-row tile
  gemm_wmma_kernel<<<16, 32, 0, stream>>>(feats, wpad, label_b, logp);
  // 8) softmax * survive, argmax
  finish_kernel<<<1, 256, 0, stream>>>(logp, out + OUT_SURV,
                                       out + OUT_LOGITS, out + OUT_LABELS);
}